// Embedding_9431748182692
// MI455X (gfx1250) — compile-verified
//
#include <hip/hip_runtime.h>
#include <hip/hip_bf16.h>

// LoRA embedding: out[b,s,:] = weight[x[b,s],:] + SCALE * (down_w.T[x[b,s],:] @ up_w.T)
// Memory-bound (~257 MB @ 23.3 TB/s); rank-8 update done with V_WMMA_F32_16X16X4_F32.
// This revision hoists all 64-bit address arithmetic out of the tile loop so the
// inner loop is pure {immediate-offset VMEM clauses + 2 WMMA + dual-FMAC epilogue}.

#define VOCAB 128000
#define DIM   1024
#define RANK  8
#define SCALE 0.125f

#define WAVES_PER_BLOCK  4
#define TOKENS_PER_WAVE  16
#define TOKENS_PER_BLOCK (WAVES_PER_BLOCK * TOKENS_PER_WAVE)   // 64
#define COL_CHUNKS       4
#define COLS_PER_CHUNK   (DIM / COL_CHUNKS)                    // 256
#define NTILES_PER_CHUNK (COLS_PER_CHUNK / 16)                 // 16

typedef __attribute__((ext_vector_type(2))) float v2f;
typedef __attribute__((ext_vector_type(8))) float v8f;

__global__ __launch_bounds__(TOKENS_PER_BLOCK * 2) void lora_embed_wmma(
    const int*   __restrict__ x,        // token indices [n_tokens]
    const float* __restrict__ weight,   // [VOCAB, DIM]
    const float* __restrict__ down_w,   // [RANK, VOCAB]
    const float* __restrict__ up_w,     // [DIM, RANK]
    float*       __restrict__ out,      // [n_tokens, DIM]
    int n_tokens)
{
    const int lane = threadIdx.x & 31;
    const int wave = threadIdx.x >> 5;
    const int tile = blockIdx.x * WAVES_PER_BLOCK + wave;      // 16-token tile id
    const long tokenBase = (long)tile * TOKENS_PER_WAVE;
    if (tokenBase >= n_tokens) return;                         // wave-uniform: EXEC stays all-1s

    const int m    = lane & 15;   // row (A) / column (B,C) within tile
    const int half = lane >> 4;
    const int k0   = half * 2;    // lanes 0-15 carry K=0,1(,4,5); lanes 16-31 carry K=2,3(,6,7)

    // Token index for this lane's A-row (lane and lane+16 hold the same token)
    const int myIdx = x[tokenBase + m];

    // --- A fragments: 16x4 f32 per WMMA; two fragments cover rank 8 ---
    // ISA layout: VGPR0 = K=k0, VGPR1 = K=k0+1 (per half-wave)
    v2f a0, a1;
    a0.x = down_w[(size_t)(k0 + 0) * VOCAB + myIdx];
    a0.y = down_w[(size_t)(k0 + 1) * VOCAB + myIdx];
    a1.x = down_w[(size_t)(k0 + 4) * VOCAB + myIdx];
    a1.y = down_w[(size_t)(k0 + 5) * VOCAB + myIdx];

    const int colChunk = blockIdx.y * COLS_PER_CHUNK;

    // Hoisted per-lane base pointers (all inner-loop addressing via immediate offsets):
    //   gather rows for the 8 token-rows this lane's C registers cover
    //   (C layout: VGPR i, lanes 0-15 -> M=i, lanes 16-31 -> M=i+8; N = lane&15)
    const float* wrow[8];
#pragma unroll
    for (int i = 0; i < 8; ++i) {
        int t = __shfl(myIdx, i + half * 8, 32);   // wave32 broadcast of token index
        wrow[i] = weight + (size_t)t * DIM + colChunk + m;
    }
    //   B-fragment base: up_w row of this lane's column; per-tile stride = 16*RANK floats
    const float* uwBase = up_w + (size_t)(colChunk + m) * RANK;
    //   output base: row half*8 of this tile at this lane's column
    float* obase = out + (size_t)(tokenBase + half * 8) * DIM + colChunk + m;

#pragma unroll 4
    for (int nt = 0; nt < NTILES_PER_CHUNK; ++nt) {
        // --- B fragments (4x16): B[k][n] = up_w[col*RANK + k], vectorized b64 loads ---
        const float* uw = uwBase + nt * (16 * RANK);
        const v2f b0 = *(const v2f*)(uw + k0);        // K = k0, k0+1
        const v2f b1 = *(const v2f*)(uw + k0 + 4);    // K = k0+4, k0+5

        // --- rank-8 LoRA tile: two f32 WMMAs, K=0..3 then K=4..7 ---
        v8f c = {0.f, 0.f, 0.f, 0.f, 0.f, 0.f, 0.f, 0.f};
        c = __builtin_amdgcn_wmma_f32_16x16x4_f32(false, a0, false, b0,
                                                  (short)0, c, false, false);
        c = __builtin_amdgcn_wmma_f32_16x16x4_f32(false, a1, false, b1,
                                                  (short)0, c, false, false);

        // --- fused epilogue along C layout: gather base embedding, scale, store ---
#pragma unroll
        for (int i = 0; i < 8; ++i) {
            const float y = wrow[i][nt * 16];                  // immediate offset nt*64B
            obase[(size_t)i * DIM + nt * 16] = fmaf(c[i], SCALE, y);
        }
    }
}

extern "C" void kernel_launch(void* const* d_in, const int* in_sizes, int n_in,
                              void* d_out, int out_size, void* d_ws, size_t ws_size,
                              hipStream_t stream) {
    const int*   x      = (const int*)  d_in[0];   // [B*S] token ids
    const float* weight = (const float*)d_in[1];   // [VOCAB, DIM]
    const float* down_w = (const float*)d_in[2];   // [RANK, VOCAB]
    const float* up_w   = (const float*)d_in[3];   // [DIM, RANK]
    float*       out    = (float*)d_out;           // [B*S, DIM]

    const int n_tokens = in_sizes[0];              // 8 * 4096 = 32768

    dim3 grid((n_tokens + TOKENS_PER_BLOCK - 1) / TOKENS_PER_BLOCK, COL_CHUNKS);
    dim3 block(TOKENS_PER_BLOCK * 2);              // 128 threads = 4 waves (wave32)

    lora_embed_wmma<<<grid, block, 0, stream>>>(x, weight, down_w, up_w, out, n_tokens);
}